// GINE_NET_56891136803148
// MI455X (gfx1250) — compile-verified
//
#include <hip/hip_runtime.h>
#include <math.h>

#define N_NODES 100000
#define N_EDGES 600000
#define FEAT 128
#define HID 256
#define CLS 8

typedef __attribute__((ext_vector_type(2))) float v2f;
typedef __attribute__((ext_vector_type(8))) float v8f;

// ---------------------------------------------------------------------------
// Initialize aggregation buffer with the self term: agg = (1+eps)*x = x.
// ---------------------------------------------------------------------------
__global__ void copy_f4_kernel(const float4* __restrict__ src,
                               float4* __restrict__ dst, int n4) {
    int i = blockIdx.x * blockDim.x + threadIdx.x;
    if (i < n4) dst[i] = src[i];
}

// ---------------------------------------------------------------------------
// Scatter-add of gathered rows: agg[dst[e]] += feat[src[e]], F floats/row.
// One thread handles 4 consecutive features of one edge (float4 gather +
// 4 native f32 global atomics). F is a power of two.
// ---------------------------------------------------------------------------
template <int F, int CHUNK_SHIFT>  // CHUNK_SHIFT = log2(F/4)
__global__ void scatter_add_kernel(const float* __restrict__ feat,
                                   const int* __restrict__ srcIdx,
                                   const int* __restrict__ dstIdx,
                                   float* __restrict__ agg) {
    const int tid = blockIdx.x * blockDim.x + threadIdx.x;
    const int e = tid >> CHUNK_SHIFT;
    if (e >= N_EDGES) return;
    const int c = (tid & ((1 << CHUNK_SHIFT) - 1)) << 2;
    const int s = srcIdx[e];
    const int d = dstIdx[e];
    const float4 v = *(const float4*)(feat + (size_t)s * F + c);
    float* p = agg + (size_t)d * F + c;
    __hip_atomic_fetch_add(p + 0, v.x, __ATOMIC_RELAXED, __HIP_MEMORY_SCOPE_AGENT);
    __hip_atomic_fetch_add(p + 1, v.y, __ATOMIC_RELAXED, __HIP_MEMORY_SCOPE_AGENT);
    __hip_atomic_fetch_add(p + 2, v.z, __ATOMIC_RELAXED, __HIP_MEMORY_SCOPE_AGENT);
    __hip_atomic_fetch_add(p + 3, v.w, __ATOMIC_RELAXED, __HIP_MEMORY_SCOPE_AGENT);
}

// ---------------------------------------------------------------------------
// GEMM1 + bias + ReLU: H[100000,256] = relu(A[100000,128] @ W[128,256] + b)
// One wave32 computes one 16x16 tile via V_WMMA_F32_16X16X4_F32.
// A-frag (16x4 f32): lanes 0-15 rows M=0..15; lane>>4 selects K pair.
// B-frag (4x16 f32): VGPR0 = K row {0|2}, VGPR1 = K row {1|3}, col = lane&15.
// C/D (16x16 f32):   VGPR r -> row r (lanes 0-15) / r+8 (lanes 16-31).
// ---------------------------------------------------------------------------
__global__ void gemm1_relu_kernel(const float* __restrict__ A,
                                  const float* __restrict__ W,
                                  const float* __restrict__ bias,
                                  float* __restrict__ H) {
    const int wave = threadIdx.x >> 5;
    const int lane = threadIdx.x & 31;
    const int tile = blockIdx.x * 8 + wave;   // 100000 tiles exactly
    const int tm = tile >> 4;                 // 6250 M tiles
    const int tn = tile & 15;                 // 16 N tiles
    const int row  = (tm << 4) + (lane & 15);
    const int col  = (tn << 4) + (lane & 15);
    const int kgrp = (lane >> 4) << 1;        // 0 or 2

    const float* arow = A + (size_t)row * FEAT + kgrp;
    const float* wcol = W + (size_t)kgrp * HID + col;

    v8f c = {};
#pragma unroll
    for (int k = 0; k < FEAT; k += 4) {
        v2f a; a.x = arow[k];                 a.y = arow[k + 1];
        v2f b; b.x = wcol[(size_t)k * HID];   b.y = wcol[(size_t)(k + 1) * HID];
        c = __builtin_amdgcn_wmma_f32_16x16x4_f32(false, a, false, b,
                                                  (short)0, c, false, false);
    }

    const float bb = bias[col];
    const int r0 = (tm << 4) + ((lane >> 4) << 3);
    float* out = H + (size_t)r0 * HID + col;
#pragma unroll
    for (int r = 0; r < 8; ++r) {
        float v = c[r] + bb;
        out[(size_t)r * HID] = v > 0.f ? v : 0.f;
    }
}

// ---------------------------------------------------------------------------
// GEMM2 + bias: OUT[100000,8] = A[100000,256] @ W2[256,8] + b2.
// N padded to 16 with a zero-padded copy of W2 in LDS (uniform reads so EXEC
// stays all-1s around every WMMA). Only the 8 valid columns are stored.
// ---------------------------------------------------------------------------
__global__ void gemm2_kernel(const float* __restrict__ A,
                             const float* __restrict__ W2,
                             const float* __restrict__ bias,
                             float* __restrict__ OUT) {
    __shared__ float wlds[HID * 16];
    for (int i = threadIdx.x; i < HID * 16; i += blockDim.x) {
        const int k = i >> 4, cc = i & 15;
        wlds[i] = (cc < CLS) ? W2[k * CLS + cc] : 0.f;
    }
    __syncthreads();

    const int wave = threadIdx.x >> 5;
    const int lane = threadIdx.x & 31;
    const int tm = blockIdx.x * 8 + wave;     // 6250 M tiles
    if (tm >= N_NODES / 16) return;           // wave-uniform exit

    const int col  = lane & 15;
    const int row  = (tm << 4) + col;
    const int kgrp = (lane >> 4) << 1;

    const float* arow = A + (size_t)row * HID + kgrp;
    const float* wc = wlds + kgrp * 16 + col;

    v8f c = {};
#pragma unroll 8
    for (int k = 0; k < HID; k += 4) {
        v2f a; a.x = arow[k];      a.y = arow[k + 1];
        v2f b; b.x = wc[k * 16];   b.y = wc[(k + 1) * 16];
        c = __builtin_amdgcn_wmma_f32_16x16x4_f32(false, a, false, b,
                                                  (short)0, c, false, false);
    }

    if (col < CLS) {
        const float bb = bias[col];
        const int r0 = (tm << 4) + ((lane >> 4) << 3);
        float* out = OUT + (size_t)r0 * CLS + col;
#pragma unroll
        for (int r = 0; r < 8; ++r) out[(size_t)r * CLS] = c[r] + bb;
    }
}

// ---------------------------------------------------------------------------
// In-place row-wise log_softmax over 8 classes.
// ---------------------------------------------------------------------------
__global__ void logsoftmax8_kernel(float* __restrict__ OUT) {
    const int r = blockIdx.x * blockDim.x + threadIdx.x;
    if (r >= N_NODES) return;
    float* p = OUT + (size_t)r * CLS;
    float v[CLS];
    float m = -1e30f;
#pragma unroll
    for (int i = 0; i < CLS; ++i) { v[i] = p[i]; m = fmaxf(m, v[i]); }
    float s = 0.f;
#pragma unroll
    for (int i = 0; i < CLS; ++i) s += expf(v[i] - m);
    const float lse = logf(s) + m;
#pragma unroll
    for (int i = 0; i < CLS; ++i) p[i] = v[i] - lse;
}

extern "C" void kernel_launch(void* const* d_in, const int* in_sizes, int n_in,
                              void* d_out, int out_size, void* d_ws, size_t ws_size,
                              hipStream_t stream) {
    (void)in_sizes; (void)n_in; (void)out_size; (void)ws_size;

    const float* x  = (const float*)d_in[0];
    const int*   ei = (const int*)d_in[1];     // [2, N_EDGES]
    const float* W1 = (const float*)d_in[2];
    const float* b1 = (const float*)d_in[3];
    const float* W2 = (const float*)d_in[4];
    const float* b2 = (const float*)d_in[5];
    const int* src = ei;
    const int* dst = ei + N_EDGES;

    char* ws = (char*)d_ws;
    float* agg1 = (float*)ws;                                        // 100000*128 f32
    float* h1   = (float*)(ws + (size_t)N_NODES * FEAT * 4);         // 100000*256 f32
    float* agg2 = (float*)(ws + (size_t)N_NODES * (FEAT + HID) * 4); // 100000*256 f32
    float* out  = (float*)d_out;                                     // 100000*8  f32

    // ---- Layer 1 ----
    {   // agg1 = x (self term), then agg1 += scatter(x[src] -> dst)
        const int n4 = N_NODES * FEAT / 4;
        copy_f4_kernel<<<(n4 + 255) / 256, 256, 0, stream>>>(
            (const float4*)x, (float4*)agg1, n4);
        const int nthreads = N_EDGES * (FEAT / 4);
        scatter_add_kernel<FEAT, 5><<<(nthreads + 255) / 256, 256, 0, stream>>>(
            x, src, dst, agg1);
        gemm1_relu_kernel<<<(N_NODES / 16) * 16 / 8, 256, 0, stream>>>(
            agg1, W1, b1, h1);
    }

    // ---- Layer 2 ----
    {   // agg2 = h1, then agg2 += scatter(h1[src] -> dst)
        const int n4 = N_NODES * HID / 4;
        copy_f4_kernel<<<(n4 + 255) / 256, 256, 0, stream>>>(
            (const float4*)h1, (float4*)agg2, n4);
        const int nthreads = N_EDGES * (HID / 4);
        scatter_add_kernel<HID, 6><<<(nthreads + 255) / 256, 256, 0, stream>>>(
            h1, src, dst, agg2);
        gemm2_kernel<<<(N_NODES / 16 + 7) / 8, 256, 0, stream>>>(
            agg2, W2, b2, out);
    }

    // ---- log_softmax (in place on d_out) ----
    logsoftmax8_kernel<<<(N_NODES + 255) / 256, 256, 0, stream>>>(out);
}